// TransitivityLoss_18631568130669
// MI455X (gfx1250) — compile-verified
//
#include <hip/hip_runtime.h>
#include <math.h>

// CDNA5 (gfx1250) wave32. FP32 WMMA 16x16x4 used for batched dot products.
typedef float v2f __attribute__((ext_vector_type(2)));
typedef float v8f __attribute__((ext_vector_type(8)));

static constexpr int D = 512;   // embedding dim (reference)

__device__ __forceinline__ float diag_sel(v8f c, int v) {
    // unrolled select: pick c[v] for v in 0..7 with compile-time lanes
    float x = c[0];
    x = (v == 1) ? c[1] : x;
    x = (v == 2) ? c[2] : x;
    x = (v == 3) ? c[3] : x;
    x = (v == 4) ? c[4] : x;
    x = (v == 5) ? c[5] : x;
    x = (v == 6) ? c[6] : x;
    x = (v == 7) ? c[7] : x;
    return x;
}

__device__ __forceinline__ float sigmoidf(float x) {
    return 1.0f / (1.0f + expf(-x));
}

// One wave (32 threads) per tile of 16 samples.
// Computes raw dots dot(e_i,e_j), dot(e_j,e_k), dot(e_i,e_k) for 16 samples
// via V_WMMA_F32_16X16X4_F32 accumulation over K=512 (diagonal of 16x16 tile),
// plus per-row sum-of-squares for the L2 normalization, then the fused
// sigmoid/violation math. Writes one partial sum per tile.
__global__ __launch_bounds__(32) void transitivity_tile_kernel(
    const float* __restrict__ emb,
    const int*  __restrict__ i_idx,
    const int*  __restrict__ j_idx,
    const int*  __restrict__ k_idx,
    float*      __restrict__ partials,
    int S)
{
    __shared__ float sd_ij[16];
    __shared__ float sd_jk[16];
    __shared__ float sd_ik[16];

    const int lane = threadIdx.x;          // 0..31
    const int tile = blockIdx.x;
    const int r    = lane & 15;            // sample-in-tile / matrix row
    const int half = lane >> 4;            // 0: K%4 in {0,1}, 1: K%4 in {2,3}

    const int s  = tile * 16 + r;
    const int sc = (s < S) ? s : (S - 1);  // clamped (keeps EXEC all-1s)

    const int ia = i_idx[sc];
    const int ja = j_idx[sc];
    const int ka = k_idx[sc];

    const float2* rowi = reinterpret_cast<const float2*>(emb + (size_t)ia * D);
    const float2* rowj = reinterpret_cast<const float2*>(emb + (size_t)ja * D);
    const float2* rowk = reinterpret_cast<const float2*>(emb + (size_t)ka * D);

    v8f c_ij = {};
    v8f c_jk = {};
    v8f c_ik = {};
    float ssq_i = 0.0f, ssq_j = 0.0f, ssq_k = 0.0f;

    // K loop: each step consumes K=4. Per-lane A/B fragment is the aligned
    // float2 at element offset k0 + 2*half of row (lane&15) -- identical
    // pattern for the A (16x4) and B (4x16) operand layouts on gfx1250.
#pragma unroll 4
    for (int k0 = 0; k0 < D; k0 += 4) {
        const int f = (k0 >> 1) + half;        // float2 index
        const float2 xi = rowi[f];
        const float2 xj = rowj[f];
        const float2 xk = rowk[f];

        v2f a_i; a_i[0] = xi.x; a_i[1] = xi.y;
        v2f a_j; a_j[0] = xj.x; a_j[1] = xj.y;
        v2f a_k; a_k[0] = xk.x; a_k[1] = xk.y;

        // D = A x B + C  (fp32 MACs, RNE -- matches reference precision)
        c_ij = __builtin_amdgcn_wmma_f32_16x16x4_f32(
                   false, a_i, false, a_j, (short)0, c_ij, false, false);
        c_jk = __builtin_amdgcn_wmma_f32_16x16x4_f32(
                   false, a_j, false, a_k, (short)0, c_jk, false, false);
        c_ik = __builtin_amdgcn_wmma_f32_16x16x4_f32(
                   false, a_i, false, a_k, (short)0, c_ik, false, false);

        // per-lane half-row sum of squares (other half is in lane^16)
        ssq_i = fmaf(xi.x, xi.x, fmaf(xi.y, xi.y, ssq_i));
        ssq_j = fmaf(xj.x, xj.x, fmaf(xj.y, xj.y, ssq_j));
        ssq_k = fmaf(xk.x, xk.x, fmaf(xk.y, xk.y, ssq_k));
    }

    // Combine half-row sums: full sumsq of row r = lane + lane^16 partials.
    ssq_i += __shfl_xor(ssq_i, 16, 32);
    ssq_j += __shfl_xor(ssq_j, 16, 32);
    ssq_k += __shfl_xor(ssq_k, 16, 32);

    // Diagonal extraction: diag(m) lives at (VGPR m, lane m) for m<8 and
    // (VGPR m-8, lane m+16) for m>=8.
    const int v = lane & 7;
    const float dij = diag_sel(c_ij, v);
    const float djk = diag_sel(c_jk, v);
    const float dik = diag_sel(c_ik, v);
    if (lane < 8) {
        sd_ij[lane] = dij; sd_jk[lane] = djk; sd_ik[lane] = dik;
    } else if (lane >= 24) {
        sd_ij[lane - 16] = dij; sd_jk[lane - 16] = djk; sd_ik[lane - 16] = dik;
    }
    __syncthreads();

    float viol = 0.0f;
    if (lane < 16 && s < S) {
        const float inv_i = 1.0f / fmaxf(sqrtf(ssq_i), 1e-12f);
        const float inv_j = 1.0f / fmaxf(sqrtf(ssq_j), 1e-12f);
        const float inv_k = 1.0f / fmaxf(sqrtf(ssq_k), 1e-12f);

        const float dot_ij = sd_ij[lane] * inv_i * inv_j;  // TEMPERATURE = 1
        const float dot_jk = sd_jk[lane] * inv_j * inv_k;
        const float dot_ik = sd_ik[lane] * inv_i * inv_k;

        const float p_ij = sigmoidf(dot_ij);
        const float p_jk = sigmoidf(dot_jk);
        const float p_ik = sigmoidf(dot_ik);

        viol = fmaxf(0.0f, p_ij * p_jk - p_ik);
    }

    // wave32 sum (lanes >=16 contribute 0)
    viol += __shfl_xor(viol, 16, 32);
    viol += __shfl_xor(viol, 8, 32);
    viol += __shfl_xor(viol, 4, 32);
    viol += __shfl_xor(viol, 2, 32);
    viol += __shfl_xor(viol, 1, 32);
    if (lane == 0) partials[tile] = viol;
}

// Deterministic single-block reduction of tile partials -> mean.
__global__ __launch_bounds__(128) void transitivity_reduce_kernel(
    const float* __restrict__ partials, float* __restrict__ out,
    int numTiles, float invS)
{
    __shared__ float buf[128];
    const int t = threadIdx.x;
    float acc = 0.0f;
    for (int idx = t; idx < numTiles; idx += 128) acc += partials[idx];
    buf[t] = acc;
    __syncthreads();
    for (int step = 64; step > 0; step >>= 1) {
        if (t < step) buf[t] += buf[t + step];
        __syncthreads();
    }
    if (t == 0) out[0] = buf[0] * invS;
}

extern "C" void kernel_launch(void* const* d_in, const int* in_sizes, int n_in,
                              void* d_out, int out_size, void* d_ws, size_t ws_size,
                              hipStream_t stream) {
    const float* emb   = (const float*)d_in[0];
    const int*   i_idx = (const int*)d_in[1];
    const int*   j_idx = (const int*)d_in[2];
    const int*   k_idx = (const int*)d_in[3];
    float*       out   = (float*)d_out;
    float*       partials = (float*)d_ws;

    const int S = in_sizes[1];
    const int numTiles = (S + 15) / 16;

    transitivity_tile_kernel<<<numTiles, 32, 0, stream>>>(
        emb, i_idx, j_idx, k_idx, partials, S);
    transitivity_reduce_kernel<<<1, 128, 0, stream>>>(
        partials, out, numTiles, 1.0f / (float)S);
}